// KitNET_AD_41695542509898
// MI455X (gfx1250) — compile-verified
//
#include <hip/hip_runtime.h>
#include <math.h>

// Problem constants (from reference)
#define NF   400          // features per row
#define NA   25           // autoencoders
#define KF   16           // visible units per AE
#define NH   12           // hidden units per AE
#define EPSV 1e-6f

#define THREADS      128  // 4 waves (wave32); each wave owns 16 rows
#define ROWS_PER_TILE 64  // 4 waves x 16 rows, double buffered
#define GRID         512
#define XSTRIDE      404  // padded LDS row stride (floats): conflict-free 16-lane
                          // column reads; XSTRIDE*4 % 16 == 0 keeps b128 alignment
#define WSTR         17   // padded W row stride
#define HSTR         20   // padded hidden row stride
#define LOADS_PER_THREAD 50  // 64 rows * 100 float4 / 128 threads

typedef float v2f __attribute__((ext_vector_type(2)));
typedef float v8f __attribute__((ext_vector_type(8)));
typedef int   v4i __attribute__((ext_vector_type(4)));

#define ASG __attribute__((address_space(1)))
#define ASL __attribute__((address_space(3)))

#if __has_builtin(__builtin_amdgcn_global_load_async_to_lds_b128)
#define ASYNC_LD_B128(g, l) \
    __builtin_amdgcn_global_load_async_to_lds_b128((ASG v4i*)(g), (ASL v4i*)(l), 0, 0)
#else
#define ASYNC_LD_B128(g, l) \
    asm volatile("global_load_async_to_lds_b128 %0, %1" \
                 :: "v"((unsigned)(unsigned long long)(ASL void*)(l)), \
                    "v"((const void*)(g)) : "memory")
#endif

#if __has_builtin(__builtin_amdgcn_s_wait_asynccnt)
#define WAIT_ASYNC0() __builtin_amdgcn_s_wait_asynccnt(0)
#else
#define WAIT_ASYNC0() asm volatile("s_wait_asynccnt 0x0" ::: "memory")
#endif

__device__ __forceinline__ float fast_sigmoid(float z)
{
    return __builtin_amdgcn_rcpf(1.0f + __expf(-z));   // v_exp + v_rcp, no div expansion
}

__global__ __launch_bounds__(THREADS, 1)
void kitnet_ad_kernel(const float* __restrict__ x,
                      const float* __restrict__ W,
                      const float* __restrict__ hb,
                      const float* __restrict__ vb,
                      const int*   __restrict__ idx,
                      float*       __restrict__ out,
                      int tilesPerWG)
{
    __shared__ float xbuf[2][ROWS_PER_TILE * XSTRIDE]; // 2 x 103,424 B (double buffer)
    __shared__ float Wp[NA * 16 * WSTR];               // [a][k][h], h>=12 zero
    __shared__ float Wt[NA * 16 * WSTR];               // [a][h][k], h>=12 rows zero
    __shared__ float hbp[NA * 16];
    __shared__ float vbp[NA * 16];
    __shared__ int   idg[NA * KF];
    __shared__ float hid[4][16 * HSTR];                // per-wave hidden tile

    const int tid  = threadIdx.x;
    const int wave = tid >> 5;
    const int lane = tid & 31;
    const int l16  = lane & 15;
    const int hi   = lane >> 4;          // wave half: 0 or 1

    // ---- stage weights / biases / indices (zero pads first) ----
    for (int i = tid; i < NA * 16 * WSTR; i += THREADS) { Wp[i] = 0.f; Wt[i] = 0.f; }
    for (int i = tid; i < NA * 16; i += THREADS) { hbp[i] = 0.f; vbp[i] = 0.f; }
    __syncthreads();
    for (int i = tid; i < NA * KF * NH; i += THREADS) {
        int a = i / (KF * NH);
        int r = i % (KF * NH);
        int k = r / NH;
        int h = r % NH;
        float w = W[i];
        Wp[(a * 16 + k) * WSTR + h] = w;   // fwd:  B[k][h]
        Wt[(a * 16 + h) * WSTR + k] = w;   // bwd:  B[h][k] = W[k][h]
    }
    for (int i = tid; i < NA * NH; i += THREADS) hbp[(i / NH) * 16 + (i % NH)] = hb[i];
    for (int i = tid; i < NA * KF; i += THREADS) { vbp[i] = vb[i]; idg[i] = idx[i]; }

    // ---- per-thread async-copy mapping: row = tid/2, half-row of 50 float4s ----
    const int rowInTile = tid >> 1;
    const int c4base    = (tid & 1) * LOADS_PER_THREAD;     // 0 or 50
    const size_t tile0  = (size_t)blockIdx.x * tilesPerWG;  // first tile index

    const float* gRow0 = x + (tile0 * ROWS_PER_TILE + rowInTile) * NF + (size_t)c4base * 4;
    float*       lRow[2] = { &xbuf[0][rowInTile * XSTRIDE + c4base * 4],
                             &xbuf[1][rowInTile * XSTRIDE + c4base * 4] };

    // prologue: kick off the whole first tile asynchronously
#pragma unroll 1
    for (int jj = 0; jj < LOADS_PER_THREAD; ++jj)
        ASYNC_LD_B128(gRow0 + (size_t)jj * 4, lRow[0] + jj * 4);

#pragma unroll 1
    for (int t = 0; t < tilesPerWG; ++t) {
        WAIT_ASYNC0();        // my share of tile t has landed in LDS
        __syncthreads();      // everyone's share landed; also fences buf[(t+1)&1] readers

        const size_t rowBase = (tile0 + t) * ROWS_PER_TILE;
        const float* gNext   = x + ((rowBase + ROWS_PER_TILE) + rowInTile) * NF
                                 + (size_t)c4base * 4;
        float*       lNext   = lRow[(t + 1) & 1];
        const bool   more    = (t + 1) < tilesPerWG;   // wave-uniform

        float* xw = &xbuf[t & 1][(wave * 16) * XSTRIDE];
        float* hw = &hid[wave][0];

        float acc[8];
#pragma unroll
        for (int j = 0; j < 8; ++j) acc[j] = 0.f;

#pragma unroll 1
        for (int a = 0; a < NA; ++a) {
            // drip-feed next tile: 2 async b128 per AE iteration (25*2 = 50 = full share)
            if (more) {
                ASYNC_LD_B128(gNext + (size_t)(2 * a) * 4,     lNext + (2 * a) * 4);
                ASYNC_LD_B128(gNext + (size_t)(2 * a + 1) * 4, lNext + (2 * a + 1) * 4);
            }

            const int abase = a * 16;

            // ---- forward: hidden_pre[16x16] = xi[16x16] @ W[16x16pad] ----
            v8f c0 = {0.f, 0.f, 0.f, 0.f, 0.f, 0.f, 0.f, 0.f};
#pragma unroll
            for (int c = 0; c < 4; ++c) {
                const int k0 = 4 * c + 2 * hi;               // K slot for VGPR0
                v2f af, bf;
                af.x = xw[l16 * XSTRIDE + idg[abase + k0]];
                af.y = xw[l16 * XSTRIDE + idg[abase + k0 + 1]];
                bf.x = Wp[(abase + k0) * WSTR + l16];
                bf.y = Wp[(abase + k0 + 1) * WSTR + l16];
                c0 = __builtin_amdgcn_wmma_f32_16x16x4_f32(false, af, false, bf,
                                                           (short)0, c0, false, false);
            }

            // bias + sigmoid; transpose hidden (lane=h -> lane=row) via LDS
            const float hbv = hbp[abase + l16];
#pragma unroll
            for (int j = 0; j < 8; ++j)
                hw[(j + 8 * hi) * HSTR + l16] = fast_sigmoid(c0[j] + hbv);
            __asm__ volatile("s_wait_dscnt 0" ::: "memory");

            // ---- backward: rec_pre[16x16] = hidden[16x12pad] @ W^T ----
            // chunk c=3 (h in [12,16)) skipped: those Wt rows are zero.
            v8f c1 = {0.f, 0.f, 0.f, 0.f, 0.f, 0.f, 0.f, 0.f};
#pragma unroll
            for (int c = 0; c < 3; ++c) {
                const int h0 = 4 * c + 2 * hi;
                v2f af, bf;
                af.x = hw[l16 * HSTR + h0];
                af.y = hw[l16 * HSTR + h0 + 1];
                bf.x = Wt[(abase + h0) * WSTR + l16];
                bf.y = Wt[(abase + h0 + 1) * WSTR + l16];
                c1 = __builtin_amdgcn_wmma_f32_16x16x4_f32(false, af, false, bf,
                                                           (short)0, c1, false, false);
            }

            // bias + sigmoid + squared error; butterfly-reduce over k (16 lanes)
            const float vbv   = vbp[abase + l16];
            const int   mycol = idg[abase + l16];
#pragma unroll
            for (int j = 0; j < 8; ++j) {
                const int row = j + 8 * hi;
                float xiv = xw[row * XSTRIDE + mycol];
                float d   = xiv - fast_sigmoid(c1[j] + vbv);
                float s   = d * d;
                s += __shfl_xor(s, 1, 32);
                s += __shfl_xor(s, 2, 32);
                s += __shfl_xor(s, 4, 32);
                s += __shfl_xor(s, 8, 32);
                acc[j] += __builtin_amdgcn_sqrtf(s * (1.0f / 16.0f) + EPSV);
            }
        }

        // ---- write out: 16 rows per wave; all 16 lanes of a half hold the sum ----
        if (l16 == 0) {
#pragma unroll
            for (int j = 0; j < 8; ++j)
                out[rowBase + wave * 16 + j + 8 * hi] = acc[j];
        }
    }
}

extern "C" void kernel_launch(void* const* d_in, const int* in_sizes, int n_in,
                              void* d_out, int out_size, void* d_ws, size_t ws_size,
                              hipStream_t stream)
{
    const float* x   = (const float*)d_in[0];
    const float* W   = (const float*)d_in[1];
    const float* hb  = (const float*)d_in[2];
    const float* vb  = (const float*)d_in[3];
    const int*   idx = (const int*)d_in[4];
    float*       out = (float*)d_out;

    const int Brows      = in_sizes[0] / NF;            // 131072
    const int totalTiles = Brows / ROWS_PER_TILE;       // 2048
    const int tilesPerWG = totalTiles / GRID;           // 4

    hipLaunchKernelGGL(kitnet_ad_kernel, dim3(GRID), dim3(THREADS), 0, stream,
                       x, W, hb, vb, idx, out, tilesPerWG);
}